// DTNAttention_23295902613826
// MI455X (gfx1250) — compile-verified
//
#include <hip/hip_runtime.h>
#include <hip/hip_bf16.h>

typedef __attribute__((ext_vector_type(2))) float v2f;
typedef __attribute__((ext_vector_type(4))) float v4f;
typedef __attribute__((ext_vector_type(8))) float v8f;

#define D_MODEL 512
#define NUM_HEADS 8
#define D_K 64
#define B_SZ 2
#define L_SEQ 256
#define BL (B_SZ * L_SEQ)   // 512
#define SCALE 0.125f        // D_K^-0.5

__device__ __forceinline__ v8f wmma4(v2f a, v2f b, v8f c) {
    // V_WMMA_F32_16X16X4_F32 : D = A(16x4, f32) x B(4x16, f32) + C(16x16, f32)
    return __builtin_amdgcn_wmma_f32_16x16x4_f32(false, a, false, b, (short)0, c,
                                                 false, false);
}

// CDNA5 async memory->LDS copy, 16B per lane, tracked by ASYNCcnt (ISA 15.18.3
// op 98). vdst = LDS byte address, vaddr = 64-bit global address (GV mode).
__device__ __forceinline__ void async_copy16(void* lds_ptr, const float* gptr) {
    unsigned lds_addr = (unsigned)(unsigned long long)lds_ptr; // low 32b = LDS offset
    asm volatile("global_load_async_to_lds_b128 %0, %1, off"
                 :: "v"(lds_addr), "v"(gptr) : "memory");
}
__device__ __forceinline__ void wait_async_all() {
    asm volatile("s_wait_asynccnt 0x0" ::: "memory");
}

// ---------------------------------------------------------------------------
// Tropical (max-plus) linear: Y[r,o] = max_i( x[r,i] + W[o,i] ) + b[o]
// grid (8 o-tiles of 64, 32 r-tiles of 16, z: 0=Q 1=K), block 256
// Tiles staged via GLOBAL_LOAD_ASYNC_TO_LDS_B128 (no VGPR round-trip).
// ---------------------------------------------------------------------------
__global__ void __launch_bounds__(256) k_trop(
        const float* __restrict__ x,
        const float* __restrict__ qW, const float* __restrict__ qb,
        const float* __restrict__ kW, const float* __restrict__ kb,
        float* __restrict__ Qo, float* __restrict__ Ko) {
    const float* W  = blockIdx.z ? kW : qW;
    const float* bs = blockIdx.z ? kb : qb;
    float*       Y  = blockIdx.z ? Ko : Qo;

    __shared__ __align__(16) float xs[16][68];
    __shared__ __align__(16) float ws[64][68];

    const int tid = threadIdx.x;
    const int r0 = blockIdx.y * 16, o0 = blockIdx.x * 64;
    const int tr = tid >> 4, tc = tid & 15;

    float a0 = -__builtin_inff(), a1 = -__builtin_inff();
    float a2 = -__builtin_inff(), a3 = -__builtin_inff();

    for (int i0 = 0; i0 < D_MODEL; i0 += 64) {
        {   // 16x64 x-tile: one 16B async copy per thread
            int r = tid >> 4, c4 = (tid & 15) << 2;
            async_copy16(&xs[r][c4], &x[(r0 + r) * D_MODEL + i0 + c4]);
        }
        #pragma unroll
        for (int t = 0; t < 4; ++t) {   // 64x64 W-tile: 4 x 16B per thread
            int idx = tid + t * 256;
            int o = idx >> 4, c4 = (idx & 15) << 2;
            async_copy16(&ws[o][c4], &W[(o0 + o) * D_MODEL + i0 + c4]);
        }
        if (i0 + 64 < D_MODEL)   // global_prefetch_b8 of next W chunk
            __builtin_prefetch(&W[(o0 + (tid >> 2)) * D_MODEL + i0 + 64], 0, 1);
        wait_async_all();
        __syncthreads();
        #pragma unroll 4
        for (int ii = 0; ii < 64; ++ii) {
            float xv = xs[tr][ii];
            a0 = fmaxf(a0, xv + ws[tc      ][ii]);
            a1 = fmaxf(a1, xv + ws[tc + 16][ii]);
            a2 = fmaxf(a2, xv + ws[tc + 32][ii]);
            a3 = fmaxf(a3, xv + ws[tc + 48][ii]);
        }
        __syncthreads();
    }
    float* yr = Y + (r0 + tr) * D_MODEL + o0;
    yr[tc     ] = a0 + bs[o0 + tc     ];
    yr[tc + 16] = a1 + bs[o0 + tc + 16];
    yr[tc + 32] = a2 + bs[o0 + tc + 32];
    yr[tc + 48] = a3 + bs[o0 + tc + 48];
}

// ---------------------------------------------------------------------------
// Y = X * W^T + b   (512x512x512, f32 WMMA).  8 waves/block, 1 tile/wave.
// grid 128, block 256
// ---------------------------------------------------------------------------
__global__ void __launch_bounds__(256) k_gemm_xwT(
        const float* __restrict__ X, const float* __restrict__ W,
        const float* __restrict__ bias, float* __restrict__ Y) {
    const int lane = threadIdx.x & 31;
    const int wave = threadIdx.x >> 5;
    const int gw = blockIdx.x * 8 + wave;          // 1024 tiles
    const int m0 = (gw >> 5) << 4;
    const int n0 = (gw & 31) << 4;
    const int m  = lane & 15;                      // M row / N col for this lane
    const int hi = lane >> 4;
    const int kh = hi << 1;                        // K half: 0 or 2

    const float* arow = X + (m0 + m) * D_MODEL + kh;
    const float* brow = W + (n0 + m) * D_MODEL + kh;   // B[k][n] = W[n][k]

    v8f c = {};
    #pragma unroll 4
    for (int k0 = 0; k0 < D_MODEL; k0 += 4) {
        v2f a = *(const v2f*)(arow + k0);
        v2f b = *(const v2f*)(brow + k0);
        c = wmma4(a, b, c);
    }
    #pragma unroll
    for (int j = 0; j < 8; ++j) {
        int row = m0 + j + 8 * hi;
        Y[row * D_MODEL + n0 + m] = c[j] + bias[n0 + m];
    }
}

// ---------------------------------------------------------------------------
// gate: g[row,h] = sigmoid( dot(x[row,:], gW[h,:]) + gb[h] )  grid 16, block 256
// ---------------------------------------------------------------------------
__global__ void __launch_bounds__(256) k_gate(
        const float* __restrict__ x, const float* __restrict__ gW,
        const float* __restrict__ gb, float* __restrict__ g) {
    int gid = blockIdx.x * 256 + threadIdx.x;      // 4096 = BL * H
    int row = gid >> 3, h = gid & 7;
    const float* xr = x + row * D_MODEL;
    const float* wr = gW + h * D_MODEL;
    float acc = 0.f;
    #pragma unroll 4
    for (int i = 0; i < D_MODEL; ++i) acc = fmaf(xr[i], wr[i], acc);
    float t = acc + gb[h];
    g[gid] = 1.f / (1.f + __expf(-t));
}

// ---------------------------------------------------------------------------
// scores = SCALE * ( g * tropQK + (1-g) * classQK )  per (b,h), 256x256
// class via WMMA (K=dk=64), trop via VALU, both from async-LDS staged tiles.
// grid (32, 16): x -> {qtile[0..16), kgroup[0..2)}, y -> {b,h}; block 256 (8 waves)
// ---------------------------------------------------------------------------
__global__ void __launch_bounds__(256) k_scores(
        const float* __restrict__ Qf, const float* __restrict__ Kf,
        const float* __restrict__ g, float* __restrict__ scores) {
    const int qt = blockIdx.x >> 1, kg = blockIdx.x & 1;
    const int b  = blockIdx.y >> 3, h  = blockIdx.y & 7;

    __shared__ __align__(16) float qs[16][68];
    __shared__ __align__(16) float ks[8][16][68];

    const int tid = threadIdx.x;
    const int q0 = qt * 16;
    {   // Q tile 16x64: one 16B async copy per thread
        int r = tid >> 4, c4 = (tid & 15) << 2;
        async_copy16(&qs[r][c4],
                     &Qf[(b * L_SEQ + q0 + r) * D_MODEL + h * D_K + c4]);
    }
    #pragma unroll
    for (int t = 0; t < 8; ++t) {   // 8 K tiles of 16x64: 8 x 16B per thread
        int idx = tid + t * 256;                 // float4 index [0,2048)
        int w = idx >> 8, r = (idx >> 4) & 15, c4 = (idx & 15) << 2;
        int kk = kg * 128 + w * 16 + r;
        async_copy16(&ks[w][r][c4],
                     &Kf[(b * L_SEQ + kk) * D_MODEL + h * D_K + c4]);
    }
    wait_async_all();
    __syncthreads();

    const int lane = tid & 31, wave = tid >> 5;
    const int m = lane & 15, hi = lane >> 4, kh = hi << 1;
    const float (*kw)[68] = ks[wave];

    // classical scores: 16 x V_WMMA_F32_16X16X4_F32
    v8f c = {};
    #pragma unroll
    for (int d0 = 0; d0 < D_K; d0 += 4) {
        v2f a, bb;
        a[0]  = qs[m][d0 + kh];  a[1]  = qs[m][d0 + kh + 1];
        bb[0] = kw[m][d0 + kh];  bb[1] = kw[m][d0 + kh + 1];
        c = wmma4(a, bb, c);
    }

    // tropical scores: max_d( Q[q,d] + K[kk,d] )  on VALU
    float trp[8];
    #pragma unroll
    for (int j = 0; j < 8; ++j) trp[j] = -__builtin_inff();
    #pragma unroll 2
    for (int d = 0; d < D_K; ++d) {
        float kv = kw[m][d];
        #pragma unroll
        for (int j = 0; j < 8; ++j)
            trp[j] = fmaxf(trp[j], qs[j + 8 * hi][d] + kv);
    }

    const int kk0 = kg * 128 + wave * 16;
    #pragma unroll
    for (int j = 0; j < 8; ++j) {
        int q = q0 + j + 8 * hi;
        float gv = g[(b * L_SEQ + q) * NUM_HEADS + h];
        float s = SCALE * (gv * trp[j] + (1.f - gv) * c[j]);
        scores[((b * NUM_HEADS + h) * L_SEQ + q) * L_SEQ + kk0 + m] = s;
    }
}

// ---------------------------------------------------------------------------
// per-row softmax with per-head temperature. grid 4096 rows, block 256
// ---------------------------------------------------------------------------
__global__ void __launch_bounds__(256) k_softmax(
        const float* __restrict__ scores, const float* __restrict__ temp,
        float* __restrict__ attn) {
    __shared__ float red[256];
    const int tid = threadIdx.x;
    const int row = blockIdx.x;
    const int h = (row >> 8) & 7;
    float v = scores[row * 256 + tid] * (1.f / temp[h]);
    red[tid] = v; __syncthreads();
    for (int s = 128; s > 0; s >>= 1) {
        if (tid < s) red[tid] = fmaxf(red[tid], red[tid + s]);
        __syncthreads();
    }
    float mx = red[0]; __syncthreads();
    float e = __expf(v - mx);
    red[tid] = e; __syncthreads();
    for (int s = 128; s > 0; s >>= 1) {
        if (tid < s) red[tid] += red[tid + s];
        __syncthreads();
    }
    attn[row * 256 + tid] = e / red[0];
}

// ---------------------------------------------------------------------------
// ctx[b,q, h*64+d] = sum_kk attn[b,h,q,kk] * V[b,kk, h*64+d]   (WMMA, K=256)
// grid 128, block 256 (8 waves); 1024 waves = 16 bh * 64 tiles
// ---------------------------------------------------------------------------
__global__ void __launch_bounds__(256) k_ctx(
        const float* __restrict__ attn, const float* __restrict__ V,
        float* __restrict__ ctx) {
    const int lane = threadIdx.x & 31, wave = threadIdx.x >> 5;
    const int gw = blockIdx.x * 8 + wave;
    const int bh = gw >> 6;
    const int tile = gw & 63;
    const int qt = tile >> 2, dt = tile & 3;
    const int b = bh >> 3, h = bh & 7;
    const int m0 = qt << 4, n0 = dt << 4;
    const int m = lane & 15, hi = lane >> 4, kh = hi << 1;

    const float* arow = attn + (bh * L_SEQ + m0 + m) * L_SEQ + kh;
    const float* bcol = V + (b * L_SEQ) * D_MODEL + h * D_K + n0 + m;

    v8f c = {};
    #pragma unroll 4
    for (int k0 = 0; k0 < L_SEQ; k0 += 4) {
        v2f a = *(const v2f*)(arow + k0);
        v2f bb;
        bb[0] = bcol[(k0 + kh)     * D_MODEL];
        bb[1] = bcol[(k0 + kh + 1) * D_MODEL];
        c = wmma4(a, bb, c);
    }
    #pragma unroll
    for (int j = 0; j < 8; ++j) {
        int q = m0 + j + 8 * hi;
        ctx[(b * L_SEQ + q) * D_MODEL + h * D_K + n0 + m] = c[j];
    }
}

// ---------------------------------------------------------------------------
extern "C" void kernel_launch(void* const* d_in, const int* in_sizes, int n_in,
                              void* d_out, int out_size, void* d_ws, size_t ws_size,
                              hipStream_t stream) {
    const float* x    = (const float*)d_in[0];
    const float* qW   = (const float*)d_in[1];
    const float* qb   = (const float*)d_in[2];
    const float* kW   = (const float*)d_in[3];
    const float* kb   = (const float*)d_in[4];
    const float* vW   = (const float*)d_in[5];
    const float* vb   = (const float*)d_in[6];
    const float* oW   = (const float*)d_in[7];
    const float* ob   = (const float*)d_in[8];
    const float* gW   = (const float*)d_in[9];
    const float* gb   = (const float*)d_in[10];
    const float* temp = (const float*)d_in[11];

    float* ws = (float*)d_ws;
    float* Qf = ws;                                       // 512*512
    float* Kf = Qf + BL * D_MODEL;                        // 512*512
    float* Vf = Kf + BL * D_MODEL;                        // 512*512
    float* gf = Vf + BL * D_MODEL;                        // 512*8
    float* sc = gf + BL * NUM_HEADS;                      // 2*8*256*256
    float* cx = sc + B_SZ * NUM_HEADS * L_SEQ * L_SEQ;    // 512*512
    // total ws: ~8.4 MB

    float* outp = (float*)d_out;                          // (B,L,D) = 262144
    float* attn = outp + BL * D_MODEL;                    // (B,H,L,L) = 1048576

    k_trop    <<<dim3(8, 32, 2), 256, 0, stream>>>(x, qW, qb, kW, kb, Qf, Kf);
    k_gemm_xwT<<<128,            256, 0, stream>>>(x, vW, vb, Vf);
    k_gate    <<<16,             256, 0, stream>>>(x, gW, gb, gf);
    k_scores  <<<dim3(32, 16),   256, 0, stream>>>(Qf, Kf, gf, sc);
    k_softmax <<<4096,           256, 0, stream>>>(sc, temp, attn);
    k_ctx     <<<128,            256, 0, stream>>>(attn, Vf, cx);
    k_gemm_xwT<<<128,            256, 0, stream>>>(cx, oW, ob, outp);
}